// SpatioTemporalGNN_86105504350430
// MI455X (gfx1250) — compile-verified
//
#include <hip/hip_runtime.h>
#include <cstdint>

// ---------------------------------------------------------------------------
// GAT (2 layers) + MLP head for CDNA5 / gfx1250 (wave32, WMMA).
// GEMMs use V_WMMA_F32_16X16X4_F32 (full fp32 precision, 16x16 C tiles).
// Edge softmax uses 3 streaming passes with HW float atomics.
// ---------------------------------------------------------------------------

typedef float v2f __attribute__((ext_vector_type(2)));
typedef float v8f __attribute__((ext_vector_type(8)));

#define DEVFN __device__ __forceinline__

DEVFN v8f wmma_f32_16x16x4(v2f a, v2f b, v8f c) {
  // 8-arg form: (neg_a, A, neg_b, B, c_mod, C, reuse_a, reuse_b)
  return __builtin_amdgcn_wmma_f32_16x16x4_f32(false, a, false, b, (short)0, c,
                                               false, false);
}

DEVFN float lrelu(float x) { return x > 0.f ? x : 0.2f * x; }
// Monotone float <-> signed-int encoding so atomicMax(int) == float max.
DEVFN int fenc(float f) {
  int i = __float_as_int(f);
  return i >= 0 ? i : (i ^ 0x7fffffff);
}
DEVFN float fdec(int i) { return __int_as_float(i >= 0 ? i : (i ^ 0x7fffffff)); }

// ---------------------------------------------------------------------------
// Tiled fp32 WMMA GEMM:  C[n x MOUT] = A[n x K] @ W[K x MOUT]  (+bias, relu)
// One block = 16 rows, MOUT/16 waves (one 16x16 tile per wave).
// W and the 16xK A-slab are staged in LDS with padded strides.
// ---------------------------------------------------------------------------
template <int K, int MOUT, bool BIAS, bool RELU>
__global__ __launch_bounds__(MOUT * 2) void gemm_wmma_kernel(
    const float* __restrict__ A, const float* __restrict__ W,
    const float* __restrict__ bias, float* __restrict__ C, int nrows) {
  constexpr int SWS = MOUT + 4;  // padded strides -> no LDS bank conflicts
  constexpr int SAS = K + 4;
  __shared__ float sW[K * SWS];
  __shared__ float sA[16 * SAS];

  const int tid = threadIdx.x;
  constexpr int NT = MOUT * 2;
  const int rowBase = blockIdx.x * 16;

  for (int i = tid; i < K * MOUT; i += NT) {
    int r = i / MOUT, c = i - r * MOUT;
    sW[r * SWS + c] = W[i];
  }
  for (int i = tid; i < 16 * K; i += NT) {
    int r = i / K, c = i - r * K;
    int gr = rowBase + r;
    sA[r * SAS + c] = (gr < nrows) ? A[(size_t)gr * K + c] : 0.0f;
  }
  __syncthreads();

  const int wave = tid >> 5;
  const int lane = tid & 31;
  const int half = lane >> 4;  // K-pair selector (lanes 16-31 hold K+2,K+3)
  const int l16 = lane & 15;   // A: row.  B/C: column.
  const int colBase = wave * 16;

  v8f acc = {0.f, 0.f, 0.f, 0.f, 0.f, 0.f, 0.f, 0.f};
#pragma unroll
  for (int k0 = 0; k0 < K; k0 += 4) {
    const int ka = k0 + 2 * half;
    v2f a, b;
    a[0] = sA[l16 * SAS + ka];
    a[1] = sA[l16 * SAS + ka + 1];
    b[0] = sW[ka * SWS + colBase + l16];
    b[1] = sW[(ka + 1) * SWS + colBase + l16];
    acc = wmma_f32_16x16x4(a, b, acc);
  }

#pragma unroll
  for (int i = 0; i < 8; ++i) {
    int r = rowBase + i + 8 * half;  // VGPR i: lanes0-15 -> M=i, 16-31 -> M=i+8
    if (r < nrows) {
      int c = colBase + l16;
      float v = acc[i];
      if (BIAS) v += bias[c];
      if (RELU) v = fmaxf(v, 0.0f);
      C[(size_t)r * MOUT + c] = v;
    }
  }
}

// ---------------------------------------------------------------------------
// Per-(node,head) attention logit dot products: es = <h, a_s>, ed = <h, a_d>
// ---------------------------------------------------------------------------
template <int H>
__global__ void attn_logits_kernel(const float* __restrict__ Hmat,
                                   const float* __restrict__ a_s,
                                   const float* __restrict__ a_d,
                                   float* __restrict__ es,
                                   float* __restrict__ ed, int n) {
  long long t = (long long)blockIdx.x * blockDim.x + threadIdx.x;
  if (t >= (long long)n * H) return;
  const int h = (int)(t & (H - 1));
  const float4* hp = (const float4*)(Hmat + (size_t)t * 64);  // t = node*H + h
  const float4* ap = (const float4*)(a_s + h * 64);
  const float4* bp = (const float4*)(a_d + h * 64);
  float s = 0.f, d = 0.f;
#pragma unroll
  for (int i = 0; i < 16; ++i) {
    float4 hv = hp[i], av = ap[i], bv = bp[i];
    s += hv.x * av.x + hv.y * av.y + hv.z * av.z + hv.w * av.w;
    d += hv.x * bv.x + hv.y * bv.y + hv.z * bv.z + hv.w * bv.w;
  }
  es[t] = s;
  ed[t] = d;
}

// ---------------------------------------------------------------------------
// Edge-state init: zero agg + denom, m = INT_MIN (below any fenc(logit)).
// ---------------------------------------------------------------------------
__global__ void init_state_kernel(float* __restrict__ agg, long long aggN,
                                  float* __restrict__ denom,
                                  int* __restrict__ m, int dn) {
  long long t = (long long)blockIdx.x * blockDim.x + threadIdx.x;
  if (t < aggN) agg[t] = 0.0f;
  if (t < dn) {
    denom[t] = 0.0f;
    m[t] = (int)0x80000000;
  }
}

// ---------------------------------------------------------------------------
// Edge pass 1: segment max of leaky_relu(es[src]+ed[dst]) via ordered-int max.
// Edges >= nE are the implicit self-loops (src = dst = e - nE).
// ---------------------------------------------------------------------------
template <int H>
__global__ void edge_max_kernel(const int* __restrict__ srcI,
                                const int* __restrict__ dstI, int nE, int nTot,
                                const float* __restrict__ es,
                                const float* __restrict__ ed,
                                int* __restrict__ m) {
  long long t = (long long)blockIdx.x * blockDim.x + threadIdx.x;
  if (t >= (long long)nTot * H) return;
  const int e = (int)(t >> (H == 2 ? 1 : 0));
  const int h = (int)(t & (H - 1));
  const int s = (e < nE) ? srcI[e] : (e - nE);
  const int d = (e < nE) ? dstI[e] : (e - nE);
  float v = lrelu(es[s * H + h] + ed[d * H + h]);
  atomicMax(&m[d * H + h], fenc(v));
}

// ---------------------------------------------------------------------------
// Edge pass 2: denom[dst] += exp(logit - max[dst])
// ---------------------------------------------------------------------------
template <int H>
__global__ void edge_denom_kernel(const int* __restrict__ srcI,
                                  const int* __restrict__ dstI, int nE,
                                  int nTot, const float* __restrict__ es,
                                  const float* __restrict__ ed,
                                  const int* __restrict__ m,
                                  float* __restrict__ denom) {
  long long t = (long long)blockIdx.x * blockDim.x + threadIdx.x;
  if (t >= (long long)nTot * H) return;
  const int e = (int)(t >> (H == 2 ? 1 : 0));
  const int h = (int)(t & (H - 1));
  const int s = (e < nE) ? srcI[e] : (e - nE);
  const int d = (e < nE) ? dstI[e] : (e - nE);
  float v = lrelu(es[s * H + h] + ed[d * H + h]);
  atomicAdd(&denom[d * H + h], __expf(v - fdec(m[d * H + h])));
}

// ---------------------------------------------------------------------------
// Edge pass 3: agg[dst, c] += alpha * H[src, c]. Thread per (edge, channel):
// channels are the fast dimension -> fully coalesced gather + atomic scatter.
// ---------------------------------------------------------------------------
template <int H>
__global__ void edge_aggregate_kernel(
    const int* __restrict__ srcI, const int* __restrict__ dstI, int nE,
    int nTot, const float* __restrict__ es, const float* __restrict__ ed,
    const int* __restrict__ m, const float* __restrict__ denom,
    const float* __restrict__ Hmat, float* __restrict__ agg) {
  constexpr int CH = H * 64;
  long long t = (long long)blockIdx.x * blockDim.x + threadIdx.x;
  if (t >= (long long)nTot * CH) return;
  const int e = (int)(t / CH);
  const int c = (int)(t & (CH - 1));
  const int h = c >> 6;
  const int s = (e < nE) ? srcI[e] : (e - nE);
  const int d = (e < nE) ? dstI[e] : (e - nE);
  float v = lrelu(es[s * H + h] + ed[d * H + h]);
  float alpha = __expf(v - fdec(m[d * H + h])) / denom[d * H + h];
  atomicAdd(&agg[(size_t)d * CH + c], alpha * Hmat[(size_t)s * CH + c]);
}

// ---------------------------------------------------------------------------
// out = elu(agg + bias)
// ---------------------------------------------------------------------------
template <int CH>
__global__ void bias_elu_kernel(const float* __restrict__ agg,
                                const float* __restrict__ b,
                                float* __restrict__ out, long long total) {
  long long t = (long long)blockIdx.x * blockDim.x + threadIdx.x;
  if (t >= total) return;
  float v = agg[t] + b[(int)(t & (CH - 1))];
  out[t] = v > 0.f ? v : expm1f(v);
}

// ---------------------------------------------------------------------------
// Final 32->1 projection.
// ---------------------------------------------------------------------------
__global__ void out_head_kernel(const float* __restrict__ fc2,
                                const float* __restrict__ ow,
                                const float* __restrict__ ob,
                                float* __restrict__ out, int n) {
  int t = blockIdx.x * blockDim.x + threadIdx.x;
  if (t >= n) return;
  const float4* p = (const float4*)(fc2 + (size_t)t * 32);
  const float4* w = (const float4*)ow;
  float acc = 0.f;
#pragma unroll
  for (int i = 0; i < 8; ++i) {
    float4 a = p[i], b = w[i];
    acc += a.x * b.x + a.y * b.y + a.z * b.z + a.w * b.w;
  }
  out[t] = acc + ob[0];
}

// ---------------------------------------------------------------------------
static inline unsigned cdiv(long long a, long long b) {
  return (unsigned)((a + b - 1) / b);
}

extern "C" void kernel_launch(void* const* d_in, const int* in_sizes, int n_in,
                              void* d_out, int out_size, void* d_ws,
                              size_t ws_size, hipStream_t stream) {
  const float* x = (const float*)d_in[0];
  const int* ei = (const int*)d_in[1];
  const float* W1 = (const float*)d_in[2];
  const float* a1s = (const float*)d_in[3];
  const float* a1d = (const float*)d_in[4];
  const float* b1 = (const float*)d_in[5];
  const float* W2 = (const float*)d_in[6];
  const float* a2s = (const float*)d_in[7];
  const float* a2d = (const float*)d_in[8];
  const float* b2 = (const float*)d_in[9];
  const float* fc1w = (const float*)d_in[10];
  const float* fc1b = (const float*)d_in[11];
  const float* fc2w = (const float*)d_in[12];
  const float* fc2b = (const float*)d_in[13];
  const float* ow = (const float*)d_in[14];
  const float* ob = (const float*)d_in[15];
  float* out = (float*)d_out;

  const int n = in_sizes[0] / 64;   // 100000
  const int nE = in_sizes[1] / 2;   // 1600000
  const int nTot = nE + n;          // + self loops
  const int* srcI = ei;
  const int* dstI = ei + nE;

  // Workspace carve-up (16-float aligned), with reuse:
  float* ws = (float*)d_ws;
  size_t off = 0;
  auto alloc = [&](size_t cnt) -> float* {
    float* p = ws + off;
    off += (cnt + 15) & ~(size_t)15;
    return p;
  };
  float* H1 = alloc((size_t)n * 128);    // H1, then elu output (in place)
  float* agg1 = alloc((size_t)n * 128);  // agg1, later fc1/fc2 outputs
  float* H2 = alloc((size_t)n * 64);     // H2, then elu output (in place)
  float* agg2 = alloc((size_t)n * 64);
  float* es1 = alloc((size_t)2 * n);
  float* ed1 = alloc((size_t)2 * n);
  int* m1 = (int*)alloc((size_t)2 * n);
  float* dn1 = alloc((size_t)2 * n);
  float* es2 = alloc((size_t)n);
  float* ed2 = alloc((size_t)n);
  int* m2 = (int*)alloc((size_t)n);
  float* dn2 = alloc((size_t)n);
  float* fc1o = agg1;                       // agg1 dead after layer-1 ELU
  float* fc2o = agg1 + (size_t)n * 64;

  const int T = 256;

  // ----- Layer 1: H1 = x @ W1 (WMMA), logits, softmax passes, ELU ----------
  gemm_wmma_kernel<64, 128, false, false>
      <<<cdiv(n, 16), 256, 0, stream>>>(x, W1, nullptr, H1, n);
  attn_logits_kernel<2>
      <<<cdiv((long long)2 * n, T), T, 0, stream>>>(H1, a1s, a1d, es1, ed1, n);
  init_state_kernel<<<cdiv((long long)n * 128, T), T, 0, stream>>>(
      agg1, (long long)n * 128, dn1, m1, 2 * n);
  edge_max_kernel<2><<<cdiv((long long)nTot * 2, T), T, 0, stream>>>(
      srcI, dstI, nE, nTot, es1, ed1, m1);
  edge_denom_kernel<2><<<cdiv((long long)nTot * 2, T), T, 0, stream>>>(
      srcI, dstI, nE, nTot, es1, ed1, m1, dn1);
  edge_aggregate_kernel<2><<<cdiv((long long)nTot * 128, T), T, 0, stream>>>(
      srcI, dstI, nE, nTot, es1, ed1, m1, dn1, H1, agg1);
  bias_elu_kernel<128><<<cdiv((long long)n * 128, T), T, 0, stream>>>(
      agg1, b1, H1, (long long)n * 128);

  // ----- Layer 2: H2 = h1 @ W2 (WMMA), 1 head ------------------------------
  gemm_wmma_kernel<128, 64, false, false>
      <<<cdiv(n, 16), 128, 0, stream>>>(H1, W2, nullptr, H2, n);
  attn_logits_kernel<1>
      <<<cdiv(n, T), T, 0, stream>>>(H2, a2s, a2d, es2, ed2, n);
  init_state_kernel<<<cdiv((long long)n * 64, T), T, 0, stream>>>(
      agg2, (long long)n * 64, dn2, m2, n);
  edge_max_kernel<1><<<cdiv(nTot, T), T, 0, stream>>>(srcI, dstI, nE, nTot,
                                                      es2, ed2, m2);
  edge_denom_kernel<1><<<cdiv(nTot, T), T, 0, stream>>>(srcI, dstI, nE, nTot,
                                                        es2, ed2, m2, dn2);
  edge_aggregate_kernel<1><<<cdiv((long long)nTot * 64, T), T, 0, stream>>>(
      srcI, dstI, nE, nTot, es2, ed2, m2, dn2, H2, agg2);
  bias_elu_kernel<64><<<cdiv((long long)n * 64, T), T, 0, stream>>>(
      agg2, b2, H2, (long long)n * 64);

  // ----- MLP head: FC1(relu) -> FC2(relu) -> 32->1 -------------------------
  gemm_wmma_kernel<64, 64, true, true>
      <<<cdiv(n, 16), 128, 0, stream>>>(H2, fc1w, fc1b, fc1o, n);
  gemm_wmma_kernel<64, 32, true, true>
      <<<cdiv(n, 16), 64, 0, stream>>>(fc1o, fc2w, fc2b, fc2o, n);
  out_head_kernel<<<cdiv(n, T), T, 0, stream>>>(fc2o, ow, ob, out, n);
}